// KPlanesFeatureField_15668040695862
// MI455X (gfx1250) — compile-verified
//
#include <hip/hip_runtime.h>

// ---------------------------------------------------------------------------
// K-Planes feature field sampler for MI455X (gfx1250, wave32).
//   out[n, 0:32]  = prod_p bilinear(planes_128[p], coords_p(n))   (C=32)
//   out[n,32:64]  = same for planes_256
//   out[n,64:96]  = same for planes_512
// Pairs: (x,y), (x,z), (y,z); grid_sample align_corners=True semantics.
//
// Strategy: planes fit in the 192MB L2. Transpose (3,C,H,W) -> (3,H,W,C) in
// d_ws so channel gathers are contiguous 128B rows; then each wave handles
// 4 points (8 lanes x float4 channels) with fully coalesced b128 gathers.
// Transpose fills LDS with CDNA5 async global->LDS copies (ASYNCcnt).
// Output stores + transpose-source reads are non-temporal so the 192MB L2
// retains only the transposed planes (132MB) that the gather stage hits.
// ---------------------------------------------------------------------------

typedef float kp_f32x4 __attribute__((ext_vector_type(4)));

#if defined(__HIP_DEVICE_COMPILE__) && __has_builtin(__builtin_amdgcn_global_load_async_to_lds_b32)
#define KP_ASYNC_LDS 1
#else
#define KP_ASYNC_LDS 0
#endif

#if KP_ASYNC_LDS
typedef __attribute__((address_space(1))) int  kp_gint;
typedef __attribute__((address_space(1))) void kp_gvoid;
typedef __attribute__((address_space(3))) int  kp_lint;
typedef __attribute__((address_space(3))) void kp_lvoid;

// cpol=1 -> TH_NT (gfx12-family CPol): read-once source data, don't pollute L2.
__device__ __forceinline__ void kp_async_g2l(const float* g, float* l) {
  kp_gint* gp = (kp_gint*)((kp_gvoid*)(void*)const_cast<float*>(g));
  kp_lint* lp = (kp_lint*)((kp_lvoid*)(void*)l);
  __builtin_amdgcn_global_load_async_to_lds_b32(gp, lp, 0, 1);
}
#endif

// ---------------------------- transpose stage ------------------------------
// in : (3, 32, H, H)  -> out : (3, H, H, 32).  blockDim = (32, 8).
// grid = (H/32, H, 3). LDS 32x32 tile (padded) for conflict-free transpose.
// H is a template arg so every stride is a compile-time immediate.
template <int H>
__global__ __launch_bounds__(256) void kp_transpose(const float* __restrict__ in,
                                                    float* __restrict__ out) {
  __shared__ float tile[32][33];
  constexpr int W   = H;
  constexpr int IMG = H * W;            // one channel image
  const int p  = blockIdx.z;
  const int y  = blockIdx.y;
  const int xb = blockIdx.x * 32;
  const int tx = threadIdx.x;
  const int ty = threadIdx.y;

  const float* src = in + ((p * 32) * H + y) * W + xb + tx;

#if KP_ASYNC_LDS
  // CDNA5 async global->LDS copy (ASYNCcnt-tracked), no VGPR staging.
#pragma unroll
  for (int k = 0; k < 4; ++k) {
    const int c = k * 8 + ty;
    kp_async_g2l(src + c * IMG, &tile[c][tx]);
  }
#if __has_builtin(__builtin_amdgcn_s_wait_asynccnt)
  __builtin_amdgcn_s_wait_asynccnt(0);
#else
  asm volatile("s_wait_asynccnt 0" ::: "memory");
#endif
#else
#pragma unroll
  for (int k = 0; k < 4; ++k) {
    const int c = k * 8 + ty;
    tile[c][tx] = __builtin_nontemporal_load(src + c * IMG);
  }
#endif
  __syncthreads();

  float* dst = out + ((p * H + y) * W + xb) * 32 + tx;  // lane = channel
#pragma unroll
  for (int k = 0; k < 4; ++k) {
    const int xx = k * 8 + ty;
    dst[xx * 32] = tile[tx][xx];      // RT store: this data must stay in L2
  }
}

// ---------------------------- sampling stage -------------------------------
// Bilinear sample of 4 consecutive channels from channel-last plane.
// Boundary: x0 = clamp(floor(ix), 0, W-2), wx = ix - x0 reproduces the
// reference's (clip + min(x0+1, W-1)) result exactly, incl. ix == W-1.
__device__ __forceinline__ kp_f32x4 kp_bil4(const float* __restrict__ base, int W,
                                            float gx, float gy, int c0) {
  const float s  = 0.5f * (float)(W - 1);
  const float ix = (gx + 1.0f) * s;
  const float iy = (gy + 1.0f) * s;
  int x0 = (int)floorf(ix);
  int y0 = (int)floorf(iy);
  x0 = x0 < 0 ? 0 : (x0 > W - 2 ? W - 2 : x0);
  y0 = y0 < 0 ? 0 : (y0 > W - 2 ? W - 2 : y0);
  const float wx = ix - (float)x0;
  const float wy = iy - (float)y0;

  const int rowS = W * 32;
  const float* r0 = base + ((y0 * W + x0) << 5) + c0;  // (y0, x0, c0)
  const kp_f32x4 v00 = *(const kp_f32x4*)(r0);
  const kp_f32x4 v01 = *(const kp_f32x4*)(r0 + 32);
  const kp_f32x4 v10 = *(const kp_f32x4*)(r0 + rowS);
  const kp_f32x4 v11 = *(const kp_f32x4*)(r0 + rowS + 32);

  const kp_f32x4 a = v00 + wx * (v01 - v00);   // contracted to v_fmac
  const kp_f32x4 b = v10 + wx * (v11 - v10);
  return a + wy * (b - a);
}

__device__ __forceinline__ kp_f32x4 kp_res4(const float* __restrict__ base, int W,
                                            float px, float py, float pz, int c0) {
  const int ps = W * W * 32;
  const kp_f32x4 a = kp_bil4(base,          W, px, py, c0);  // pair (0,1)
  const kp_f32x4 b = kp_bil4(base + ps,     W, px, pz, c0);  // pair (0,2)
  const kp_f32x4 c = kp_bil4(base + 2 * ps, W, py, pz, c0);  // pair (1,2)
  return a * b * c;
}

// blockDim = (32, 8): wave = 4 points x (8 lanes * float4 = 32 channels).
// All gathers are fully coalesced global_load_b128 against L2-resident data.
// Output is a 402MB stream -> non-temporal b128 stores (don't evict planes).
__global__ __launch_bounds__(256) void kp_sample_t(const float* __restrict__ x,
                                                   const float* __restrict__ t128,
                                                   const float* __restrict__ t256,
                                                   const float* __restrict__ t512,
                                                   float* __restrict__ out, int N) {
  const int lane = threadIdx.x;                       // 0..31
  const int n    = blockIdx.x * 32 + threadIdx.y * 4 + (lane >> 3);
  const int c0   = (lane & 7) << 2;                   // channel group: 0,4,..,28
  if (n >= N) return;

  const float px = x[3 * (size_t)n + 0];
  const float py = x[3 * (size_t)n + 1];
  const float pz = x[3 * (size_t)n + 2];
  float* o = out + (size_t)n * 96 + c0;

  __builtin_nontemporal_store(kp_res4(t128, 128, px, py, pz, c0), (kp_f32x4*)(o));
  __builtin_nontemporal_store(kp_res4(t256, 256, px, py, pz, c0), (kp_f32x4*)(o + 32));
  __builtin_nontemporal_store(kp_res4(t512, 512, px, py, pz, c0), (kp_f32x4*)(o + 64));
}

// ------------------- fallback: direct gather (no transpose) ----------------
__device__ __forceinline__ float kp_bil1(const float* __restrict__ chbase, int W,
                                         float gx, float gy) {
  const float s  = 0.5f * (float)(W - 1);
  const float ix = (gx + 1.0f) * s;
  const float iy = (gy + 1.0f) * s;
  int x0 = (int)floorf(ix);
  int y0 = (int)floorf(iy);
  x0 = x0 < 0 ? 0 : (x0 > W - 2 ? W - 2 : x0);
  y0 = y0 < 0 ? 0 : (y0 > W - 2 ? W - 2 : y0);
  const float wx = ix - (float)x0;
  const float wy = iy - (float)y0;
  const int idx = y0 * W + x0;
  const float v00 = chbase[idx];
  const float v01 = chbase[idx + 1];
  const float v10 = chbase[idx + W];
  const float v11 = chbase[idx + W + 1];
  const float a = fmaf(wx, v01 - v00, v00);
  const float b = fmaf(wx, v11 - v10, v10);
  return fmaf(wy, b - a, a);
}

__device__ __forceinline__ float kp_res1(const float* __restrict__ planes, int W,
                                         float px, float py, float pz, int c) {
  const size_t img = (size_t)W * W;
  const float* b0 = planes + (size_t)c * img;            // plane 0, channel c
  const float* b1 = b0 + 32 * img;                       // plane 1
  const float* b2 = b1 + 32 * img;                       // plane 2
  return kp_bil1(b0, W, px, py) * kp_bil1(b1, W, px, pz) * kp_bil1(b2, W, py, pz);
}

__global__ __launch_bounds__(256) void kp_sample_direct(const float* __restrict__ x,
                                                        const float* __restrict__ p128,
                                                        const float* __restrict__ p256,
                                                        const float* __restrict__ p512,
                                                        float* __restrict__ out, int N) {
  const int c = threadIdx.x;                     // channel
  const int n = blockIdx.x * 8 + threadIdx.y;    // point
  if (n >= N) return;
  const float px = x[3 * (size_t)n + 0];
  const float py = x[3 * (size_t)n + 1];
  const float pz = x[3 * (size_t)n + 2];
  float* o = out + (size_t)n * 96 + c;
  __builtin_nontemporal_store(kp_res1(p128, 128, px, py, pz, c), o);
  __builtin_nontemporal_store(kp_res1(p256, 256, px, py, pz, c), o + 32);
  __builtin_nontemporal_store(kp_res1(p512, 512, px, py, pz, c), o + 64);
}

// ------------------------------- launcher ----------------------------------
extern "C" void kernel_launch(void* const* d_in, const int* in_sizes, int n_in,
                              void* d_out, int out_size, void* d_ws, size_t ws_size,
                              hipStream_t stream) {
  const float* x    = (const float*)d_in[0];
  const float* p128 = (const float*)d_in[1];
  const float* p256 = (const float*)d_in[2];
  const float* p512 = (const float*)d_in[3];
  float* out = (float*)d_out;
  const int N = in_sizes[0] / 3;

  const size_t f128 = (size_t)3 * 32 * 128 * 128;
  const size_t f256 = (size_t)3 * 32 * 256 * 256;
  const size_t f512 = (size_t)3 * 32 * 512 * 512;
  const size_t need = (f128 + f256 + f512) * sizeof(float);

  const dim3 blk(32, 8);
  if (ws_size >= need) {
    float* t128 = (float*)d_ws;
    float* t256 = t128 + f128;
    float* t512 = t256 + f256;
    kp_transpose<128><<<dim3(128 / 32, 128, 3), blk, 0, stream>>>(p128, t128);
    kp_transpose<256><<<dim3(256 / 32, 256, 3), blk, 0, stream>>>(p256, t256);
    kp_transpose<512><<<dim3(512 / 32, 512, 3), blk, 0, stream>>>(p512, t512);
    kp_sample_t<<<(N + 31) / 32, blk, 0, stream>>>(x, t128, t256, t512, out, N);
  } else {
    kp_sample_direct<<<(N + 7) / 8, blk, 0, stream>>>(x, p128, p256, p512, out, N);
  }
}